// DrugInteractionGAT_20890720928084
// MI455X (gfx1250) — compile-verified
//
#include <hip/hip_runtime.h>
#include <math.h>

#define N_NODES 50000
#define N_EDGES 800000
#define F_IN    128
#define HID     64
#define HEADS   4
#define NPAIRS  16384

typedef float v2f __attribute__((ext_vector_type(2)));
typedef float v8f __attribute__((ext_vector_type(8)));

#define CDIV(a, b) (((a) + (b) - 1) / (b))

// ---------- order-preserving float <-> uint encoding for atomicMax ----------
__device__ __forceinline__ unsigned fenc(float f) {
  unsigned u = __float_as_uint(f);
  return (u & 0x80000000u) ? ~u : (u | 0x80000000u);
}
__device__ __forceinline__ float fdec(unsigned e) {
  unsigned u = (e & 0x80000000u) ? (e & 0x7FFFFFFFu) : ~e;
  return __uint_as_float(u);
}

// ---------- FP32 WMMA GEMM: C[M x NC] = A[M x K] @ B[K x NC] ----------
// One wave32 per 16x64 output strip: 4 accumulators share one A fragment per
// K-step -> 4x less A traffic, 4 back-to-back V_WMMA_F32_16X16X4_F32 per step.
template <int K, int NC>
__global__ void __launch_bounds__(256)
gemm_wmma_f32(const float* __restrict__ A, const float* __restrict__ B,
              float* __restrict__ C, int M) {
  static_assert(NC % 64 == 0, "NC must be a multiple of 64");
  const int gtid   = blockIdx.x * blockDim.x + threadIdx.x;
  const int wave   = gtid >> 5;
  const int lane   = gtid & 31;
  const int tilesN = NC / 64;
  const int tilesM = M / 16;
  if (wave >= tilesM * tilesN) return;
  const int tm = wave / tilesN;
  const int tn = wave % tilesN;
  const int lo = lane & 15;   // row (A) / col (B,C)
  const int hi = lane >> 4;   // selects K pair (A,B) / M half (C)

  const float* arow = A + (size_t)(tm * 16 + lo) * K;
  const float* bcol = B + (size_t)(tn * 64 + lo);
  v8f acc0 = {}, acc1 = {}, acc2 = {}, acc3 = {};
#pragma unroll 8
  for (int k0 = 0; k0 < K; k0 += 4) {
    // A 16x4: lanes 0-15 hold K=k0,k0+1 ; lanes 16-31 hold K=k0+2,k0+3
    v2f a;
    a.x = arow[k0 + 2 * hi];
    a.y = arow[k0 + 2 * hi + 1];
    // B 4x16 fragments for 4 adjacent column tiles
    const float* b0 = bcol + (size_t)(k0 + 2 * hi) * NC;
    const float* b1 = bcol + (size_t)(k0 + 2 * hi + 1) * NC;
    v2f bb0, bb1, bb2, bb3;
    bb0.x = b0[0];  bb0.y = b1[0];
    bb1.x = b0[16]; bb1.y = b1[16];
    bb2.x = b0[32]; bb2.y = b1[32];
    bb3.x = b0[48]; bb3.y = b1[48];
    acc0 = __builtin_amdgcn_wmma_f32_16x16x4_f32(false, a, false, bb0, (short)0, acc0, false, false);
    acc1 = __builtin_amdgcn_wmma_f32_16x16x4_f32(false, a, false, bb1, (short)0, acc1, false, false);
    acc2 = __builtin_amdgcn_wmma_f32_16x16x4_f32(false, a, false, bb2, (short)0, acc2, false, false);
    acc3 = __builtin_amdgcn_wmma_f32_16x16x4_f32(false, a, false, bb3, (short)0, acc3, false, false);
  }
  // C/D: vgpr r -> M = r (lanes 0-15) / r+8 (lanes 16-31), N = lane%16
  float* crow = C + (size_t)(tm * 16 + 8 * hi) * NC + tn * 64 + lo;
#pragma unroll
  for (int r = 0; r < 8; ++r) {
    crow[(size_t)r * NC + 0]  = acc0[r];
    crow[(size_t)r * NC + 16] = acc1[r];
    crow[(size_t)r * NC + 32] = acc2[r];
    crow[(size_t)r * NC + 48] = acc3[r];
  }
}

// ---------- degree + self-loop attr (fill_value='mean') ----------
__global__ void k_deg_accum(const int* __restrict__ dst, const float* __restrict__ ea,
                            float* __restrict__ deg, float* __restrict__ lsum, int E) {
  int e = blockIdx.x * blockDim.x + threadIdx.x;
  if (e >= E) return;
  int d = dst[e];
  atomicAdd(&deg[d], 1.0f);
  atomicAdd(&lsum[d], ea[e]);
}
__global__ void k_loop_div(float* __restrict__ la, const float* __restrict__ deg, int n) {
  int i = blockIdx.x * blockDim.x + threadIdx.x;
  if (i >= n) return;
  la[i] = la[i] / fmaxf(deg[i], 1.0f);
}

// ---------- collapse (ea @ We) . att_edge  ->  ea * ce[h] ----------
__global__ void k_ce(const float* __restrict__ We1, const float* __restrict__ ae1,
                     const float* __restrict__ We2, const float* __restrict__ ae2,
                     float* __restrict__ ce1, float* __restrict__ ce2) {
  int t = threadIdx.x;
  if (t < HEADS) {
    float s = 0.f;
    for (int c = 0; c < HID; ++c) s = fmaf(We1[t * HID + c], ae1[t * HID + c], s);
    ce1[t] = s;
  } else if (t == HEADS) {
    float s = 0.f;
    for (int c = 0; c < HID; ++c) s = fmaf(We2[c], ae2[c], s);
    ce2[0] = s;
  }
}

// ---------- per-node attention logits: a_src/a_dst ----------
__global__ void k_node_attn(const float* __restrict__ h, const float* __restrict__ as_,
                            const float* __restrict__ ad_, float* __restrict__ asrc,
                            float* __restrict__ adst, int n, int heads, int C) {
  int i = blockIdx.x * blockDim.x + threadIdx.x;
  if (i >= n * heads) return;
  int hh = i % heads, nd = i / heads;
  const float* hp = h + (size_t)nd * heads * C + (size_t)hh * C;
  const float* ap = as_ + (size_t)hh * C;
  const float* bp = ad_ + (size_t)hh * C;
  float s = 0.f, d = 0.f;
  for (int c = 0; c < C; ++c) {
    float v = hp[c];
    s = fmaf(v, ap[c], s);
    d = fmaf(v, bp[c], d);
  }
  asrc[i] = s;
  adst[i] = d;
}

// ---------- pass 1: alpha = leaky_relu(...), segment max via atomicMax ----------
__global__ void k_edge_alpha(const int* __restrict__ src, const int* __restrict__ dst,
                             const float* __restrict__ ea, const float* __restrict__ loop_attr,
                             const float* __restrict__ asrc, const float* __restrict__ adst,
                             const float* __restrict__ ce, float* __restrict__ alpha,
                             unsigned* __restrict__ amax, int E, int n, int heads) {
  long i = (long)blockIdx.x * blockDim.x + threadIdx.x;
  long tot = (long)(E + n) * heads;
  if (i >= tot) return;
  int hh = (int)(i % heads);
  long e = i / heads;
  int s, d;
  float a;
  if (e < E) { s = src[e]; d = dst[e]; a = ea[e]; }
  else       { s = d = (int)(e - E);   a = loop_attr[s]; }
  float al = asrc[(size_t)s * heads + hh] + adst[(size_t)d * heads + hh] + a * ce[hh];
  al = (al > 0.f) ? al : 0.2f * al;   // negative_slope = 0.2
  alpha[i] = al;
  atomicMax(&amax[(size_t)d * heads + hh], fenc(al));
}

// ---------- pass 2: ex = exp(alpha - max), segment sum via atomicAdd ----------
__global__ void k_edge_exp(const int* __restrict__ dst, float* __restrict__ alpha_ex,
                           const unsigned* __restrict__ amax, float* __restrict__ denom,
                           int E, int n, int heads) {
  long i = (long)blockIdx.x * blockDim.x + threadIdx.x;
  long tot = (long)(E + n) * heads;
  if (i >= tot) return;
  int hh = (int)(i % heads);
  long e = i / heads;
  int d = (e < E) ? dst[e] : (int)(e - E);
  float m = fdec(amax[(size_t)d * heads + hh]);
  float ex = expf(alpha_ex[i] - m);
  alpha_ex[i] = ex;
  atomicAdd(&denom[(size_t)d * heads + hh], ex);
}

// ---------- pass 3: out[dst] += h[src] * w  (one wave32 per edge) ----------
__global__ void k_edge_agg(const int* __restrict__ src, const int* __restrict__ dst,
                           const float* __restrict__ h, const float* __restrict__ ex,
                           const float* __restrict__ denom, float* __restrict__ out,
                           int E, int n, int heads, int C) {
  long g = (long)blockIdx.x * blockDim.x + threadIdx.x;
  long wave = g >> 5;
  int lane = (int)(g & 31);
  if (wave >= (long)(E + n)) return;
  int s, d;
  if (wave < E) { s = src[wave]; d = dst[wave]; }
  else          { s = d = (int)(wave - E); }
  int F = heads * C;
  const float* hp = h + (size_t)s * F;
  float* op = out + (size_t)d * F;
  for (int c = lane; c < F; c += 32) {
    int hh = c / C;  // C==64 -> shift
    float w = ex[wave * heads + hh] / denom[(size_t)d * heads + hh];
    atomicAdd(&op[c], hp[c] * w);
  }
}

// ---------- bias + ELU (in place) ----------
__global__ void k_bias_elu(float* __restrict__ x, const float* __restrict__ b,
                           long n, int F) {
  long i = (long)blockIdx.x * blockDim.x + threadIdx.x;
  if (i >= n * F) return;
  float v = x[i] + b[i % F];
  x[i] = (v > 0.f) ? v : (expf(v) - 1.f);
}

// ---------- pair scoring: one wave32 per pair; b2 folded in ----------
__global__ void k_pair(const float* __restrict__ h2, const float* __restrict__ b2,
                       const int* __restrict__ pairs, const float* __restrict__ Wlin,
                       const float* __restrict__ blin, float* __restrict__ out, int P) {
  int g = blockIdx.x * blockDim.x + threadIdx.x;
  int wave = g >> 5, lane = g & 31;
  if (wave >= P) return;
  int i = pairs[2 * wave], j = pairs[2 * wave + 1];
  float acc = (h2[(size_t)i * HID + lane]      + b2[lane])      * Wlin[lane]
            + (h2[(size_t)i * HID + lane + 32] + b2[lane + 32]) * Wlin[lane + 32]
            + (h2[(size_t)j * HID + lane]      + b2[lane])      * Wlin[HID + lane]
            + (h2[(size_t)j * HID + lane + 32] + b2[lane + 32]) * Wlin[HID + lane + 32];
#pragma unroll
  for (int off = 16; off; off >>= 1) acc += __shfl_xor(acc, off, 32);
  if (lane == 0) out[wave] = 1.f / (1.f + expf(-(acc + blin[0])));
}

extern "C" void kernel_launch(void* const* d_in, const int* in_sizes, int n_in,
                              void* d_out, int out_size, void* d_ws, size_t ws_size,
                              hipStream_t stream) {
  const float* x     = (const float*)d_in[0];
  const int*   esrc  = (const int*)d_in[1];
  const int*   edst  = (const int*)d_in[2];
  const float* eattr = (const float*)d_in[3];
  const int*   pairs = (const int*)d_in[4];
  const float* W1    = (const float*)d_in[5];
  const float* We1   = (const float*)d_in[6];
  const float* as1   = (const float*)d_in[7];
  const float* ad1   = (const float*)d_in[8];
  const float* ae1   = (const float*)d_in[9];
  const float* b1    = (const float*)d_in[10];
  const float* W2    = (const float*)d_in[11];
  const float* We2   = (const float*)d_in[12];
  const float* as2   = (const float*)d_in[13];
  const float* ad2   = (const float*)d_in[14];
  const float* ae2   = (const float*)d_in[15];
  const float* b2    = (const float*)d_in[16];
  const float* Wlin  = (const float*)d_in[17];
  const float* blin  = (const float*)d_in[18];
  float* out = (float*)d_out;

  const int N = N_NODES, E = N_EDGES, EL = N_EDGES + N_NODES;
  const int F1 = HEADS * HID;  // 256

  // ---- carve workspace (256B-aligned slices) ----
  char* p = (char*)d_ws;
  auto carve = [&](size_t bytes) -> void* {
    void* r = (void*)p;
    p += (bytes + 255) & ~(size_t)255;
    return r;
  };
  float*    deg    = (float*)carve((size_t)N * 4);
  float*    lattr  = (float*)carve((size_t)N * 4);            // loopsum -> loop_attr
  float*    h1     = (float*)carve((size_t)N * F1 * 4);       // x @ W1
  float*    asrc1  = (float*)carve((size_t)N * HEADS * 4);
  float*    adst1  = (float*)carve((size_t)N * HEADS * 4);
  float*    ce1    = (float*)carve(256);
  float*    ce2    = (float*)carve(256);
  unsigned* amax1  = (unsigned*)carve((size_t)N * HEADS * 4);
  float*    den1   = (float*)carve((size_t)N * HEADS * 4);
  float*    ex1    = (float*)carve((size_t)EL * HEADS * 4);
  float*    agg1   = (float*)carve((size_t)N * F1 * 4);       // layer1 out -> elu in place
  float*    h2p    = (float*)carve((size_t)N * HID * 4);      // h1act @ W2
  float*    asrc2  = (float*)carve((size_t)N * 4);
  float*    adst2  = (float*)carve((size_t)N * 4);
  unsigned* amax2  = (unsigned*)carve((size_t)N * 4);
  float*    den2   = (float*)carve((size_t)N * 4);
  float*    ex2    = (float*)carve((size_t)EL * 4);
  float*    agg2   = (float*)carve((size_t)N * HID * 4);      // layer2 out (pre-bias)

  const int B = 256;

  // ---- zero accumulators (graph-capture-safe async memsets) ----
  hipMemsetAsync(deg,   0, (size_t)N * 4, stream);
  hipMemsetAsync(lattr, 0, (size_t)N * 4, stream);
  hipMemsetAsync(amax1, 0, (size_t)N * HEADS * 4, stream);  // 0 < fenc(any finite)
  hipMemsetAsync(den1,  0, (size_t)N * HEADS * 4, stream);
  hipMemsetAsync(agg1,  0, (size_t)N * F1 * 4, stream);
  hipMemsetAsync(amax2, 0, (size_t)N * 4, stream);
  hipMemsetAsync(den2,  0, (size_t)N * 4, stream);
  hipMemsetAsync(agg2,  0, (size_t)N * HID * 4, stream);

  // ---- shared precompute ----
  k_deg_accum<<<CDIV(E, B), B, 0, stream>>>(edst, eattr, deg, lattr, E);
  k_loop_div<<<CDIV(N, B), B, 0, stream>>>(lattr, deg, N);
  k_ce<<<1, 32, 0, stream>>>(We1, ae1, We2, ae2, ce1, ce2);

  // ---- layer 1 ----
  gemm_wmma_f32<F_IN, 256><<<CDIV((N / 16) * (256 / 64) * 32, B), B, 0, stream>>>(x, W1, h1, N);
  k_node_attn<<<CDIV(N * HEADS, B), B, 0, stream>>>(h1, as1, ad1, asrc1, adst1, N, HEADS, HID);
  k_edge_alpha<<<CDIV((long)EL * HEADS, B), B, 0, stream>>>(esrc, edst, eattr, lattr,
                                                            asrc1, adst1, ce1, ex1, amax1,
                                                            E, N, HEADS);
  k_edge_exp<<<CDIV((long)EL * HEADS, B), B, 0, stream>>>(edst, ex1, amax1, den1, E, N, HEADS);
  k_edge_agg<<<CDIV((long)EL * 32, B), B, 0, stream>>>(esrc, edst, h1, ex1, den1, agg1,
                                                       E, N, HEADS, HID);
  k_bias_elu<<<CDIV((long)N * F1, B), B, 0, stream>>>(agg1, b1, (long)N, F1);

  // ---- layer 2 (1 head) ----
  gemm_wmma_f32<256, HID><<<CDIV((N / 16) * (HID / 64) * 32, B), B, 0, stream>>>(agg1, W2, h2p, N);
  k_node_attn<<<CDIV(N, B), B, 0, stream>>>(h2p, as2, ad2, asrc2, adst2, N, 1, HID);
  k_edge_alpha<<<CDIV((long)EL, B), B, 0, stream>>>(esrc, edst, eattr, lattr,
                                                    asrc2, adst2, ce2, ex2, amax2, E, N, 1);
  k_edge_exp<<<CDIV((long)EL, B), B, 0, stream>>>(edst, ex2, amax2, den2, E, N, 1);
  k_edge_agg<<<CDIV((long)EL * 32, B), B, 0, stream>>>(esrc, edst, h2p, ex2, den2, agg2,
                                                       E, N, 1, HID);

  // ---- pair scoring (b2 folded in) ----
  k_pair<<<CDIV(NPAIRS * 32, B), B, 0, stream>>>(agg2, b2, pairs, Wlin, blin, out, NPAIRS);
}